// JobShopGraphConv_46712064311848
// MI455X (gfx1250) — compile-verified
//
#include <hip/hip_runtime.h>

#define N_NODES 100000
#define E_EDGES 1600000
#define D 64

typedef float v2f __attribute__((ext_vector_type(2)));
typedef float v8f __attribute__((ext_vector_type(8)));

// ---------------- utility kernels ----------------

__global__ void zero_kernel(float* __restrict__ p, int n) {
  int i = blockIdx.x * blockDim.x + threadIdx.x;
  if (i < n) p[i] = 0.0f;
}

__global__ void init_out_kernel(float* __restrict__ out, const float* __restrict__ b_fc, int n) {
  int i = blockIdx.x * blockDim.x + threadIdx.x;
  if (i < n) out[i] = b_fc[0];
}

// ---------------- edge scatter (segment_sum) ----------------
// one wave per edge: 32 lanes x float2 = 64 features.
// gather is a coalesced 256B read per edge; scatter via f32 atomics into L2.
__global__ void scatter_kernel(const float* __restrict__ feat,
                               const long long* __restrict__ src,
                               const long long* __restrict__ dst,
                               float* __restrict__ agg) {
  int tid = blockIdx.x * blockDim.x + threadIdx.x;
  int e = tid >> 5;            // edge id (grid sized exactly: no OOB check needed)
  int f = (tid & 31) << 1;     // feature pair
  int s = (int)src[e];
  int d = (int)dst[e];
  v2f v = *(const v2f*)(feat + (size_t)s * D + f);
  atomicAdd(agg + (size_t)d * D + f,     v.x);
  atomicAdd(agg + (size_t)d * D + f + 1, v.y);
}

// ---------------- fused GraphConv GEMM via fp32 WMMA ----------------
// out_tile[16x16] = agg_tile[16x64] @ Wrel[n][k]^T + root_tile[16x64] @ Wroot^T + b
// K stepped 4 at a time with V_WMMA_F32_16X16X4_F32 (full fp32 precision).
// FUSE_FC: multiply relu'd tile by W_fc column slice, 16-lane reduce, atomicAdd
// into the [N,1] output (out must be pre-initialized to b_fc).
template <bool FUSE_FC>
__global__ void gemm_relu_kernel(const float* __restrict__ agg,
                                 const float* __restrict__ rootin,
                                 const float* __restrict__ Wrel,
                                 const float* __restrict__ Wroot,
                                 const float* __restrict__ bias,
                                 float* __restrict__ hout,
                                 const float* __restrict__ Wfc,
                                 float* __restrict__ out) {
  int wave = (blockIdx.x * blockDim.x + threadIdx.x) >> 5;
  int lane = threadIdx.x & 31;
  int rowTile = wave >> 2;     // 6250 row tiles of 16 nodes
  int colTile = wave & 3;      // 4 col tiles of 16 features
  int rowbase = rowTile * 16;
  int colbase = colTile * 16;
  int n    = lane & 15;        // element-in-16 index (M for A-frag, N for B/C-frag)
  int half = lane >> 4;        // selects K pair {k0,k0+1} vs {k0+2,k0+3}
  int koff = 2 * half;

  const float* aRel  = agg    + (size_t)(rowbase + n) * D + koff;
  const float* aRoot = rootin + (size_t)(rowbase + n) * D + koff;
  const float* bRel  = Wrel   + (size_t)(colbase + n) * D + koff;  // B[k][n] = W[n][k]
  const float* bRoot = Wroot  + (size_t)(colbase + n) * D + koff;

  v8f c = {};
#pragma unroll
  for (int k0 = 0; k0 < D; k0 += 4) {
    v2f ar = *(const v2f*)(aRel + k0);
    v2f br = *(const v2f*)(bRel + k0);
    c = __builtin_amdgcn_wmma_f32_16x16x4_f32(false, ar, false, br, (short)0, c, false, false);
    v2f ax = *(const v2f*)(aRoot + k0);
    v2f bx = *(const v2f*)(bRoot + k0);
    c = __builtin_amdgcn_wmma_f32_16x16x4_f32(false, ax, false, bx, (short)0, c, false, false);
  }

  float b = bias[colbase + n];

  if (!FUSE_FC) {
    // C/D layout: VGPR r holds M=r (lanes 0-15) / M=r+8 (lanes 16-31), N=lane&15
#pragma unroll
    for (int r = 0; r < 8; ++r) {
      int m = rowbase + r + 8 * half;
      float v = c[r] + b;
      v = v > 0.0f ? v : 0.0f;
      hout[(size_t)m * D + colbase + n] = v;
    }
  } else {
    float wfc = Wfc[colbase + n];
#pragma unroll
    for (int r = 0; r < 8; ++r) {
      float v = c[r] + b;
      v = v > 0.0f ? v : 0.0f;
      float p = v * wfc;
      // reduce the 16 N-lanes of this half-wave (xor<=8 stays within the half)
      p += __shfl_xor(p, 1);
      p += __shfl_xor(p, 2);
      p += __shfl_xor(p, 4);
      p += __shfl_xor(p, 8);
      if (n == 0) atomicAdd(out + rowbase + r + 8 * half, p);
    }
  }
}

// ---------------- launch ----------------

extern "C" void kernel_launch(void* const* d_in, const int* in_sizes, int n_in,
                              void* d_out, int out_size, void* d_ws, size_t ws_size,
                              hipStream_t stream) {
  const float*     x       = (const float*)d_in[0];
  const long long* ei      = (const long long*)d_in[1];
  const float*     W1_rel  = (const float*)d_in[2];
  const float*     W1_root = (const float*)d_in[3];
  const float*     b1      = (const float*)d_in[4];
  const float*     W2_rel  = (const float*)d_in[5];
  const float*     W2_root = (const float*)d_in[6];
  const float*     b2      = (const float*)d_in[7];
  const float*     W_fc    = (const float*)d_in[8];
  const float*     b_fc    = (const float*)d_in[9];
  float*           out     = (float*)d_out;

  float* agg = (float*)d_ws;                 // [N,64]
  float* h1  = agg + (size_t)N_NODES * D;    // [N,64]

  const long long* src = ei;             // edge_index[0]
  const long long* dst = ei + E_EDGES;   // edge_index[1]

  const int featElems   = N_NODES * D;                    // 6.4M
  const int zeroBlocks  = (featElems + 255) / 256;
  const int scatBlocks  = (E_EDGES * 32) / 256;           // exact: 200000
  const int gemmBlocks  = ((N_NODES / 16) * 4) / 8;       // 25000 waves / 8 per block

  // layer 1: agg = segment_sum(x[src], dst); h1 = relu(agg@W1rel^T + x@W1root^T + b1)
  zero_kernel<<<zeroBlocks, 256, 0, stream>>>(agg, featElems);
  scatter_kernel<<<scatBlocks, 256, 0, stream>>>(x, src, dst, agg);
  gemm_relu_kernel<false><<<gemmBlocks, 256, 0, stream>>>(
      agg, x, W1_rel, W1_root, b1, h1, nullptr, nullptr);

  // layer 2 + fused fc head: out = relu(...)@Wfc^T + b_fc
  zero_kernel<<<zeroBlocks, 256, 0, stream>>>(agg, featElems);
  scatter_kernel<<<scatBlocks, 256, 0, stream>>>(h1, src, dst, agg);
  init_out_kernel<<<(N_NODES + 255) / 256, 256, 0, stream>>>(out, b_fc, N_NODES);
  gemm_relu_kernel<true><<<gemmBlocks, 256, 0, stream>>>(
      agg, h1, W2_rel, W2_root, b2, nullptr, W_fc, out);
}